// EnhancedStatistics_53008486367496
// MI455X (gfx1250) — compile-verified
//
#include <hip/hip_runtime.h>
#include <math.h>

// ---------------- problem constants ----------------
#define B_  128
#define T_  4096
#define D_  64
#define N_  (T_ * D_)        // 262144 elements per batch
#define F_  (T_ / 2 + 1)     // 2049 rfft bins
#define EPSF 1e-8f

typedef float v2f __attribute__((ext_vector_type(2)));
typedef float v8f __attribute__((ext_vector_type(8)));

// =====================================================================
// Kernel A: one block per batch. Single pass over x and mask.
// Produces: xm[b,t] (row means), obs[b,t] (mask count per time),
//           chan[b, 0..63]=col sums, chan[b,64..127]=col sumsq,
//           scal[b, 0..5] = {S1,S2,S3,S4, lag1-product-sum, mask-sum}
// =====================================================================
__global__ void kA(const float* __restrict__ x, const float* __restrict__ mask,
                   float* __restrict__ xm, float* __restrict__ obs,
                   float* __restrict__ chan, float* __restrict__ scal)
{
    const int b    = blockIdx.x;
    const int lane = threadIdx.x & 31;
    const int wave = threadIdx.x >> 5;            // 0..7
    const float* xb = x    + (size_t)b * N_;
    const float* mb = mask + (size_t)b * N_;

    float s1 = 0.f, s2 = 0.f, s3 = 0.f, s4 = 0.f, ac = 0.f, ms = 0.f;
    float cs0 = 0.f, cq0 = 0.f, cs1 = 0.f, cq1 = 0.f;   // channels lane, lane+32

    for (int t = wave; t < T_; t += 8) {
        float v0 = xb[t * D_ + lane];
        float v1 = xb[t * D_ + lane + 32];
        float m0 = mb[t * D_ + lane];
        float m1 = mb[t * D_ + lane + 32];

        s1 += v0 + v1;
        s2 += v0 * v0 + v1 * v1;
        s3 += v0 * v0 * v0 + v1 * v1 * v1;
        s4 += (v0 * v0) * (v0 * v0) + (v1 * v1) * (v1 * v1);
        ms += m0 + m1;
        cs0 += v0; cq0 += v0 * v0;
        cs1 += v1; cq1 += v1 * v1;

        if (t < T_ - 1) {
            float n0 = xb[(t + 1) * D_ + lane];
            float n1 = xb[(t + 1) * D_ + lane + 32];
            ac += v0 * n0 + v1 * n1;
        }
        // row reductions (64 values across this wave's 32 lanes x2)
        float rs = v0 + v1;
        float rm = m0 + m1;
        for (int off = 16; off; off >>= 1) {
            rs += __shfl_xor(rs, off, 32);
            rm += __shfl_xor(rm, off, 32);
        }
        if (lane == 0) {
            xm [(size_t)b * T_ + t] = rs * (1.0f / 64.0f);
            obs[(size_t)b * T_ + t] = rm;
        }
    }

    // ---- per-channel reduction across the 8 waves ----
    __shared__ float chs[8][64];
    __shared__ float chq[8][64];
    chs[wave][lane]      = cs0;  chs[wave][lane + 32] = cs1;
    chq[wave][lane]      = cq0;  chq[wave][lane + 32] = cq1;
    __syncthreads();
    if (threadIdx.x < 64) {
        float s = 0.f, q = 0.f;
        for (int w = 0; w < 8; ++w) { s += chs[w][threadIdx.x]; q += chq[w][threadIdx.x]; }
        chan[b * 128 + threadIdx.x]      = s;
        chan[b * 128 + 64 + threadIdx.x] = q;
    }

    // ---- scalar reductions ----
    __shared__ float red[8][6];
    float vals[6] = { s1, s2, s3, s4, ac, ms };
    #pragma unroll
    for (int i = 0; i < 6; ++i) {
        float v = vals[i];
        for (int off = 16; off; off >>= 1) v += __shfl_xor(v, off, 32);
        vals[i] = v;
    }
    if (lane == 0)
        for (int i = 0; i < 6; ++i) red[wave][i] = vals[i];
    __syncthreads();
    if (threadIdx.x == 0) {
        float o[6] = { 0, 0, 0, 0, 0, 0 };
        for (int w = 0; w < 8; ++w)
            for (int i = 0; i < 6; ++i) o[i] += red[w][i];
        for (int i = 0; i < 6; ++i) scal[b * 8 + i] = o[i];
    }
}

// =====================================================================
// Kernel Gram (WMMA + async-to-LDS): gram[b] = Xc^T * Xc (64x64, K=4096).
// Identity Xc^T Xc = X^T X - T * mu mu^T lets the async DMA stream RAW x;
// the rank-1 centering correction is applied once at writeback.
// 256 threads = 8 waves; each wave owns a 16(M)x32(N) strip so the A
// fragment is reused across two V_WMMA_F32_16X16X4_F32 issues.
// Double-buffered LDS: chunk k+1 streams in via
// global_load_async_to_lds_b128 (ASYNCcnt) while chunk k is consumed.
// =====================================================================
#define GP 68   // LDS row stride in floats (64 + 4 pad): 272B, 16B-aligned,
                // and conflict-free column reads (bank = (4k + d) & 63)

__device__ __forceinline__ void async_tile_load(const float* __restrict__ gbase,
                                                float* lbase, int tid)
{
#if defined(__AMDGCN__)
    // 64 rows x 64 floats as 1024 16-byte packets; 4 per thread.
    #pragma unroll
    for (int i = 0; i < 4; ++i) {
        int j   = tid + i * 256;
        int row = j >> 4;
        int grp = j & 15;
        const float* g = gbase + row * 64 + grp * 4;
        unsigned lds   = (unsigned)(size_t)(lbase + row * GP + grp * 4);
        asm volatile("global_load_async_to_lds_b128 %0, %1, off"
                     :: "v"(lds), "v"((unsigned long long)(size_t)g)
                     : "memory");
    }
#else
    (void)gbase; (void)lbase; (void)tid;
#endif
}

__device__ __forceinline__ void wait_async_lds(void)
{
#if defined(__AMDGCN__)
    asm volatile("s_wait_asynccnt 0" ::: "memory");
#endif
}

__global__ void kGram(const float* __restrict__ x, const float* __restrict__ chan,
                      float* __restrict__ gram)
{
    __shared__ __align__(16) float buf[2][64 * GP];   // double-buffered [k][d] tiles
    __shared__ float cmean[64];

    const int b    = blockIdx.x;
    const int tid  = threadIdx.x;
    const int lane = tid & 31;
    const int wave = tid >> 5;                 // 0..7
    const int mi   = (wave >> 1) * 16;         // M tile base (channel)
    const int ni   = (wave &  1) * 32;         // N strip base (channel), covers ni..ni+31

    if (tid < 64) cmean[tid] = chan[b * 128 + tid] * (1.0f / (float)T_);

    const float* xb = x + (size_t)b * N_;
    v8f acc0 = { 0.f, 0.f, 0.f, 0.f, 0.f, 0.f, 0.f, 0.f };
    v8f acc1 = { 0.f, 0.f, 0.f, 0.f, 0.f, 0.f, 0.f, 0.f };

    const int half = lane >> 4;     // 0 -> K pair (0,1); 1 -> K pair (2,3)
    const int l    = lane & 15;

    async_tile_load(xb, buf[0], tid);          // prologue: chunk 0 in flight

    for (int kb = 0; kb < T_; kb += 64) {
        const int cur = (kb >> 6) & 1;
        wait_async_lds();                      // my packets have landed
        __syncthreads();                       // everyone's packets have landed
        if (kb + 64 < T_)
            async_tile_load(xb + (size_t)(kb + 64) * 64, buf[cur ^ 1], tid);

        const float* t0 = buf[cur];
        #pragma unroll 4
        for (int k = 0; k < 64; k += 4) {
            const int kr = k + 2 * half;
            v2f a, b0, b1;
            // A: 16x4, ISA layout: lanes0-15 hold K0/K1, lanes16-31 K2/K3
            a.x  = t0[(kr + 0) * GP + (mi + l)];
            a.y  = t0[(kr + 1) * GP + (mi + l)];
            b0.x = t0[(kr + 0) * GP + (ni + l)];
            b0.y = t0[(kr + 1) * GP + (ni + l)];
            b1.x = t0[(kr + 0) * GP + (ni + 16 + l)];
            b1.y = t0[(kr + 1) * GP + (ni + 16 + l)];
            acc0 = __builtin_amdgcn_wmma_f32_16x16x4_f32(
                       false, a, false, b0, (short)0, acc0, false, false);
            acc1 = __builtin_amdgcn_wmma_f32_16x16x4_f32(
                       false, a, false, b1, (short)0, acc1, false, false);
        }
        __syncthreads();                       // all reads of buf[cur] done before refill
    }

    // C/D layout: VGPR r: lanes0-15 -> M = r, N = lane; lanes16-31 -> M = r+8
    float* g = gram + (size_t)b * 64 * 64;
    #pragma unroll
    for (int r = 0; r < 8; ++r) {
        int m  = mi + r + half * 8;
        float cmT = cmean[m] * (float)T_;
        int n0 = ni + l;
        int n1 = ni + 16 + l;
        g[m * 64 + n0] = acc0[r] - cmT * cmean[n0];
        g[m * 64 + n1] = acc1[r] - cmT * cmean[n1];
    }
}

// =====================================================================
// Kernel DFT: power spectrum of xm row -> dom_freq (col 2), spec_ent (col 3)
// Exact phase reduction: angle index p = (f*t) mod 4096.
// =====================================================================
__global__ void kDFT(const float* __restrict__ xm, float* __restrict__ out)
{
    __shared__ float s_xm[T_];
    __shared__ float s_pw[F_];
    __shared__ float rsum[256];
    __shared__ float rmax[256];
    __shared__ int   rarg[256];

    const int b   = blockIdx.x;
    const int tid = threadIdx.x;

    for (int i = tid; i < T_; i += 256) s_xm[i] = xm[(size_t)b * T_ + i];
    __syncthreads();

    const float w0 = 6.283185307179586f / (float)T_;
    for (int f = tid; f < F_; f += 256) {
        float cr = 0.f, ci = 0.f;
        int p = 0;                         // (f*t) mod 4096, exact
        for (int t = 0; t < T_; ++t) {
            float sn, cs;
            __sincosf(w0 * (float)p, &sn, &cs);
            float v = s_xm[t];
            cr += v * cs;
            ci -= v * sn;
            p += f; p &= (T_ - 1);
        }
        s_pw[f] = cr * cr + ci * ci;
    }
    __syncthreads();

    float lsum = 0.f, lmax = -1.f; int larg = 0;
    for (int f = tid; f < F_; f += 256) {
        float pw = s_pw[f];
        lsum += pw;
        if (pw > lmax) { lmax = pw; larg = f; }
    }
    rsum[tid] = lsum; rmax[tid] = lmax; rarg[tid] = larg;
    __syncthreads();
    for (int s = 128; s; s >>= 1) {
        if (tid < s) {
            rsum[tid] += rsum[tid + s];
            if (rmax[tid + s] > rmax[tid] ||
                (rmax[tid + s] == rmax[tid] && rarg[tid + s] < rarg[tid])) {
                rmax[tid] = rmax[tid + s];
                rarg[tid] = rarg[tid + s];
            }
        }
        __syncthreads();
    }
    const float S   = rsum[0];
    const int   arg = rarg[0];
    __syncthreads();

    float lent = 0.f;
    for (int f = tid; f < F_; f += 256) {
        float pn = s_pw[f] / (S + EPSF);
        lent -= pn * logf(pn + EPSF);
    }
    rsum[tid] = lent;
    __syncthreads();
    for (int s = 128; s; s >>= 1) { if (tid < s) rsum[tid] += rsum[tid + s]; __syncthreads(); }

    if (tid == 0) {
        out[b * 17 + 2] = (float)arg / (float)(T_ / 2);
        out[b * 17 + 3] = rsum[0];
    }
}

// =====================================================================
// Kernel Temporal: cols 0,1,4,5,9,10,11,12,13,14
// =====================================================================
__global__ void kTemporal(const float* __restrict__ xm, const float* __restrict__ obs,
                          const float* __restrict__ scal, float* __restrict__ out)
{
    __shared__ float s[T_];
    __shared__ float red[256];
    __shared__ int   ired[256];

    const int b   = blockIdx.x;
    const int tid = threadIdx.x;

    for (int i = tid; i < T_; i += 256) s[i] = xm[(size_t)b * T_ + i];
    __syncthreads();

    // mean of xm
    float ls = 0.f;
    for (int i = tid; i < T_; i += 256) ls += s[i];
    red[tid] = ls; __syncthreads();
    for (int st = 128; st; st >>= 1) { if (tid < st) red[tid] += red[tid + st]; __syncthreads(); }
    const float xmean = red[0] * (1.0f / (float)T_);
    __syncthreads();

    float tn = 0.f, roc = 0.f; int zc = 0, pk = 0;
    for (int t = tid; t < T_; t += 256) {
        tn += ((float)t - 2047.5f) * (s[t] - xmean);
        if (t < T_ - 1) {
            float d = s[t + 1] - s[t];
            roc += fabsf(d);
            float c0 = s[t] - xmean, c1 = s[t + 1] - xmean;
            if (c1 * c0 < 0.f) zc++;
        }
        if (t >= 1 && t < T_ - 1) {
            float d1 = s[t + 1] - s[t];
            float d0 = s[t] - s[t - 1];
            if (d1 * d0 < 0.f) pk++;
        }
    }
    red[tid] = tn; __syncthreads();
    for (int st = 128; st; st >>= 1) { if (tid < st) red[tid] += red[tid + st]; __syncthreads(); }
    const float tnum = red[0]; __syncthreads();
    red[tid] = roc; __syncthreads();
    for (int st = 128; st; st >>= 1) { if (tid < st) red[tid] += red[tid + st]; __syncthreads(); }
    const float rocsum = red[0]; __syncthreads();
    ired[tid] = zc; __syncthreads();
    for (int st = 128; st; st >>= 1) { if (tid < st) ired[tid] += ired[tid + st]; __syncthreads(); }
    const int zcAll = ired[0]; __syncthreads();
    ired[tid] = pk; __syncthreads();
    for (int st = 128; st; st >>= 1) { if (tid < st) ired[tid] += ired[tid + st]; __syncthreads(); }
    const int pkAll = ired[0]; __syncthreads();

    // ---- obs stats (reuse s) ----
    for (int i = tid; i < T_; i += 256) s[i] = obs[(size_t)b * T_ + i];
    __syncthreads();
    float ds = 0.f, dq = 0.f; int fmin = T_;
    for (int t = tid; t < T_; t += 256) {
        float o  = s[t];
        float dn = o * (1.0f / (64.0f + EPSF));
        ds += dn; dq += dn * dn;
        if (o > 0.f && t < fmin) fmin = t;
    }
    red[tid] = ds; __syncthreads();
    for (int st = 128; st; st >>= 1) { if (tid < st) red[tid] += red[tid + st]; __syncthreads(); }
    const float dsum = red[0]; __syncthreads();
    red[tid] = dq; __syncthreads();
    for (int st = 128; st; st >>= 1) { if (tid < st) red[tid] += red[tid + st]; __syncthreads(); }
    const float dqsum = red[0]; __syncthreads();
    ired[tid] = fmin; __syncthreads();
    for (int st = 128; st; st >>= 1) { if (tid < st) ired[tid] = min(ired[tid], ired[tid + st]); __syncthreads(); }
    const int firstIdx = ired[0];

    if (tid == 0) {
        const float Nf = (float)N_;
        float S1 = scal[b * 8 + 0], S2 = scal[b * 8 + 1];
        float S3 = scal[b * 8 + 2], S4 = scal[b * 8 + 3];
        float AC = scal[b * 8 + 4], MS = scal[b * 8 + 5];
        float mu = S1 / Nf;
        float m2 = S2 / Nf - mu * mu;
        float m3 = S3 / Nf - 3.f * mu * (S2 / Nf) + 2.f * mu * mu * mu;
        float m4 = S4 / Nf - 4.f * mu * (S3 / Nf) + 6.f * mu * mu * (S2 / Nf)
                   - 3.f * mu * mu * mu * mu;

        float muD  = dsum / (float)T_;
        float dvar = (dqsum - (float)T_ * muD * muD) / (float)(T_ - 1);
        float first = (firstIdx < T_) ? (float)firstIdx / (float)T_ : 0.f;

        const float tden = 5726622720.0f;   // sum (t - 2047.5)^2, t = 0..4095

        out[b * 17 + 0]  = AC / ((float)(T_ - 1) * (float)D_);
        out[b * 17 + 1]  = tnum / (tden + EPSF);
        out[b * 17 + 4]  = m3 / (powf(m2, 1.5f) + EPSF);
        out[b * 17 + 5]  = m4 / (m2 * m2 + EPSF);
        out[b * 17 + 9]  = (float)pkAll / (float)(T_ - 2);
        out[b * 17 + 10] = (float)zcAll / (float)(T_ - 1);
        out[b * 17 + 11] = rocsum / (float)(T_ - 1);
        out[b * 17 + 12] = 1.f - MS / ((float)N_ + EPSF);
        out[b * 17 + 13] = dvar;
        out[b * 17 + 14] = first;
    }
}

// =====================================================================
// Kernel Quant: histogram-selection quantiles (cols 6,7,8).
// 6 order-statistic ranks refined simultaneously, 4 rounds x 256 bins.
// =====================================================================
#define NB 256
__global__ void kQuant(const float* __restrict__ x, float* __restrict__ out)
{
    __shared__ float s_lo[6], s_hi[6];
    __shared__ int   s_base[6];
    __shared__ int   hist[6][NB];
    __shared__ float red[256];

    const int b   = blockIdx.x;
    const int tid = threadIdx.x;
    const float* xb = x + (size_t)b * N_;

    // pass 0: min / max
    float mn = 3.4e38f, mx = -3.4e38f;
    for (int i = tid; i < N_; i += 256) {
        float v = xb[i];
        mn = fminf(mn, v); mx = fmaxf(mx, v);
    }
    red[tid] = mn; __syncthreads();
    for (int s = 128; s; s >>= 1) { if (tid < s) red[tid] = fminf(red[tid], red[tid + s]); __syncthreads(); }
    const float gmn = red[0]; __syncthreads();
    red[tid] = mx; __syncthreads();
    for (int s = 128; s; s >>= 1) { if (tid < s) red[tid] = fmaxf(red[tid], red[tid + s]); __syncthreads(); }
    const float gmx = red[0]; __syncthreads();

    const int ranks[6] = { 65535, 65536, 131071, 131072, 196607, 196608 };
    if (tid < 6) {
        s_lo[tid] = gmn;
        s_hi[tid] = gmx + fmaxf((gmx - gmn) * 1e-6f, 1e-12f);
        s_base[tid] = 0;
    }
    __syncthreads();

    for (int round = 0; round < 4; ++round) {
        for (int i = tid; i < 6 * NB; i += 256) (&hist[0][0])[i] = 0;
        __syncthreads();
        for (int i = tid; i < N_; i += 256) {
            float v = xb[i];
            #pragma unroll
            for (int r = 0; r < 6; ++r) {
                float lo = s_lo[r], hi = s_hi[r];
                if (v >= lo && v < hi) {
                    float w = (hi - lo) * (1.0f / NB);
                    int bin = (int)((v - lo) / w);
                    bin = bin < 0 ? 0 : (bin > NB - 1 ? NB - 1 : bin);
                    atomicAdd(&hist[r][bin], 1);
                }
            }
        }
        __syncthreads();
        if (tid < 6) {
            int r = ranks[tid] - s_base[tid];
            int cum = 0, j = 0;
            for (j = 0; j < NB; ++j) {
                int c = hist[tid][j];
                if (cum + c > r) break;
                cum += c;
            }
            if (j >= NB) j = NB - 1;
            float lo = s_lo[tid], hi = s_hi[tid];
            float w = (hi - lo) * (1.0f / NB);
            s_lo[tid]  = lo + (float)j * w;
            s_hi[tid]  = lo + (float)(j + 1) * w;
            s_base[tid] += cum;
        }
        __syncthreads();
    }

    if (tid == 0) {
        float v0 = s_lo[0], v1 = s_lo[1];
        float v2 = s_lo[2], v3 = s_lo[3];
        float v4 = s_lo[4], v5 = s_lo[5];
        out[b * 17 + 6] = v0 + 0.75f * (v1 - v0);
        out[b * 17 + 7] = v2 + 0.50f * (v3 - v2);
        out[b * 17 + 8] = v4 + 0.25f * (v5 - v4);
    }
}

// =====================================================================
// Kernel Corr: mean off-diagonal correlation (col 15)
// =====================================================================
__global__ void kCorr(const float* __restrict__ gram, const float* __restrict__ chan,
                      float* __restrict__ out)
{
    __shared__ float sd[64];
    __shared__ float red[256];
    const int b   = blockIdx.x;
    const int tid = threadIdx.x;

    if (tid < 64) {
        float s = chan[b * 128 + tid];
        float q = chan[b * 128 + 64 + tid];
        float var = (q - s * s * (1.0f / (float)T_)) * (1.0f / (float)(T_ - 1));
        sd[tid] = sqrtf(fmaxf(var, 0.f));
    }
    __syncthreads();

    const float* g = gram + (size_t)b * 4096;
    float acc = 0.f;
    for (int i = tid; i < 4096; i += 256) {
        int r = i >> 6, c = i & 63;
        if (r != c) {
            float cov = g[i] * (1.0f / ((float)(T_ - 1) + EPSF));
            acc += cov / (sd[r] * sd[c] + EPSF);
        }
    }
    red[tid] = acc; __syncthreads();
    for (int s = 128; s; s >>= 1) { if (tid < s) red[tid] += red[tid + s]; __syncthreads(); }
    if (tid == 0)
        out[b * 17 + 15] = red[0] / ((float)D_ * (float)(D_ - 1) + EPSF);
}

// =====================================================================
// Kernel Eig: var_explained = lambda_max(gram) / trace(gram)  (col 16)
// Power iteration, 64 threads per batch.
// =====================================================================
__global__ void kEig(const float* __restrict__ gram, float* __restrict__ out)
{
    __shared__ float g[64 * 64];
    __shared__ float v[64];
    __shared__ float w[64];
    __shared__ float red[64];

    const int b   = blockIdx.x;
    const int tid = threadIdx.x;   // 64 threads

    for (int i = tid; i < 4096; i += 64) g[i] = gram[(size_t)b * 4096 + i];
    v[tid] = 0.125f;               // ||v|| = 1
    __syncthreads();

    for (int it = 0; it < 64; ++it) {
        float acc = 0.f;
        for (int j = 0; j < 64; ++j) acc += g[tid * 64 + j] * v[j];
        w[tid] = acc;
        red[tid] = acc * acc; __syncthreads();
        for (int s = 32; s; s >>= 1) { if (tid < s) red[tid] += red[tid + s]; __syncthreads(); }
        float nrm = sqrtf(red[0]) + 1e-30f;
        __syncthreads();
        v[tid] = w[tid] / nrm;
        __syncthreads();
    }

    float acc = 0.f;
    for (int j = 0; j < 64; ++j) acc += g[tid * 64 + j] * v[j];
    red[tid] = v[tid] * acc; __syncthreads();
    for (int s = 32; s; s >>= 1) { if (tid < s) red[tid] += red[tid + s]; __syncthreads(); }
    const float lam = red[0];
    __syncthreads();
    red[tid] = g[tid * 64 + tid]; __syncthreads();
    for (int s = 32; s; s >>= 1) { if (tid < s) red[tid] += red[tid + s]; __syncthreads(); }
    if (tid == 0)
        out[b * 17 + 16] = lam / red[0];
}

// =====================================================================
// Launch
// =====================================================================
extern "C" void kernel_launch(void* const* d_in, const int* in_sizes, int n_in,
                              void* d_out, int out_size, void* d_ws, size_t ws_size,
                              hipStream_t stream)
{
    (void)in_sizes; (void)n_in; (void)out_size; (void)ws_size;
    const float* x    = (const float*)d_in[0];
    const float* mask = (const float*)d_in[1];
    float* out = (float*)d_out;

    float* ws   = (float*)d_ws;
    float* xm   = ws;                       // B*T
    float* obs  = ws + (size_t)B_ * T_;     // B*T
    float* gram = ws + (size_t)2 * B_ * T_; // B*64*64
    float* chan = gram + (size_t)B_ * 64 * 64;  // B*128
    float* scal = chan + (size_t)B_ * 128;      // B*8

    kA       <<<B_, 256, 0, stream>>>(x, mask, xm, obs, chan, scal);
    kGram    <<<B_, 256, 0, stream>>>(x, chan, gram);
    kDFT     <<<B_, 256, 0, stream>>>(xm, out);
    kTemporal<<<B_, 256, 0, stream>>>(xm, obs, scal, out);
    kQuant   <<<B_, 256, 0, stream>>>(x, out);
    kCorr    <<<B_, 256, 0, stream>>>(gram, chan, out);
    kEig     <<<B_, 64, 0, stream>>>(gram, out);
}